// UnifiedEmbedding_72524817761022
// MI455X (gfx1250) — compile-verified
//
#include <hip/hip_runtime.h>
#include <stdint.h>

// xxh32 primes
#define XXP2 2246822519u
#define XXP3 3266489917u
#define XXP4 668265263u
#define XXP5 374761393u

__device__ __forceinline__ uint32_t rotl32(uint32_t v, int r) {
    return (v << r) | (v >> (32 - r));
}

// xxh32 of a single 4-byte little-endian word
__device__ __forceinline__ uint32_t xxh32_4(uint32_t val, uint32_t seed) {
    uint32_t acc = seed + XXP5 + 4u;     // seed + PRIME5 + len
    acc += val * XXP3;
    acc = rotl32(acc, 17) * XXP4;
    acc ^= acc >> 15;  acc *= XXP2;
    acc ^= acc >> 13;  acc *= XXP3;
    acc ^= acc >> 16;
    return acc;
}

// 64 f32 per embedding row = 256 bytes. One wave (32 lanes x 16B) moves 2 rows
// per async instruction; 8 groups per wave = 16 rows / 4KB per wave pass.
// Data path: global_load_async_to_lds_b128 (gather, RT so rows stay in L2)
//   -> s_wait_asynccnt 0 ->
// global_store_async_from_lds_b128 th:TH_STORE_NT (write-once stream, don't
// pollute the 192MB L2 that should hold the 256MB table). Data never touches VGPRs.
#define ROW_BYTES 256
#define GROUPS 8
#define PAIRS_PER_WAVE (GROUPS * 2)
#define WAVES_PER_BLOCK 8

template <int F_CT, uint32_t LEV_CT>
__global__ __launch_bounds__(256)
void UnifiedEmbedding_gather_async(const int* __restrict__ x,
                                   const int* __restrict__ fnum,
                                   const float* __restrict__ emb,
                                   float* __restrict__ out,
                                   int F_rt, uint32_t lev_rt, int totalPairs) {
    const int      F      = F_CT   ? F_CT   : F_rt;     // constant -> magic-number div
    const uint32_t levels = LEV_CT ? LEV_CT : lev_rt;

    __shared__ char lds[WAVES_PER_BLOCK * PAIRS_PER_WAVE * ROW_BYTES]; // 32 KB/block

    const int lane = threadIdx.x & 31;
    const int waveInBlk = threadIdx.x >> 5;
    const int globalWave = blockIdx.x * WAVES_PER_BLOCK + waveInBlk;
    const int pairBase = globalWave * PAIRS_PER_WAVE;
    if (pairBase >= totalPairs) return;  // no barriers used; safe to exit early

    // Low 32 bits of the flat address of an LDS location == hardware dsaddr
    const uint32_t waveLds =
        (uint32_t)(uintptr_t)(&lds[waveInBlk * PAIRS_PER_WAVE * ROW_BYTES]);

    uint32_t ldsOff[GROUPS];
    unsigned long long gaddr[GROUPS];
    unsigned long long oaddr[GROUPS];
    bool act[GROUPS];

    const unsigned long long embBase = (unsigned long long)(uintptr_t)emb;
    const unsigned long long outBase = (unsigned long long)(uintptr_t)out;

#pragma unroll
    for (int g = 0; g < GROUPS; ++g) {
        // lanes 0-15 -> pair 2g, lanes 16-31 -> pair 2g+1; each lane moves 16B
        const int pair = pairBase + g * 2 + (lane >> 4);
        act[g] = (pair < totalPairs);
        const int row  = pair / F;
        const int feat = pair - row * F;
        const uint32_t h   = xxh32_4((uint32_t)x[row], (uint32_t)fnum[feat]);
        const uint32_t idx = h % levels;

        gaddr[g]  = embBase + (unsigned long long)idx * ROW_BYTES
                            + (unsigned)((lane & 15) * 16);
        // output is contiguous in pair order: group base + lane*16 covers 512B
        oaddr[g]  = outBase + (unsigned long long)(pairBase + g * 2) * ROW_BYTES
                            + (unsigned)(lane * 16);
        ldsOff[g] = waveLds + (uint32_t)(g * 512 + lane * 16);
    }

    // Gather: global -> LDS (per-lane 16B, 512B/instruction, 8 in flight, ASYNCcnt)
#pragma unroll
    for (int g = 0; g < GROUPS; ++g)
        if (act[g])
            asm volatile("global_load_async_to_lds_b128 %0, %1, off"
                         :: "v"(ldsOff[g]), "v"(gaddr[g]) : "memory");

    asm volatile("s_wait_asynccnt 0" ::: "memory");

    // Stream: LDS -> global, fully coalesced, non-temporal (write-once output)
#pragma unroll
    for (int g = 0; g < GROUPS; ++g)
        if (act[g])
            asm volatile("global_store_async_from_lds_b128 %0, %1, off th:TH_STORE_NT"
                         :: "v"(oaddr[g]), "v"(ldsOff[g]) : "memory");

    asm volatile("s_wait_asynccnt 0" ::: "memory");
}

extern "C" void kernel_launch(void* const* d_in, const int* in_sizes, int n_in,
                              void* d_out, int out_size, void* d_ws, size_t ws_size,
                              hipStream_t stream) {
    const int*   x    = (const int*)d_in[0];    // [N]
    const int*   fnum = (const int*)d_in[1];    // [F]
    const float* emb  = (const float*)d_in[2];  // [levels, 64]
    float*       out  = (float*)d_out;          // [N, F*64]

    const int N = in_sizes[0];
    const int F = in_sizes[1];
    const uint32_t levels = (uint32_t)(in_sizes[2] / 64);
    const int totalPairs = N * F;

    const int waves  = (totalPairs + PAIRS_PER_WAVE - 1) / PAIRS_PER_WAVE;
    const int blocks = (waves + WAVES_PER_BLOCK - 1) / WAVES_PER_BLOCK;

    if (F == 20 && levels == 1000000u) {
        UnifiedEmbedding_gather_async<20, 1000000u>
            <<<dim3(blocks), dim3(256), 0, stream>>>(
                x, fnum, emb, out, F, levels, totalPairs);
    } else {
        UnifiedEmbedding_gather_async<0, 0u>
            <<<dim3(blocks), dim3(256), 0, stream>>>(
                x, fnum, emb, out, F, levels, totalPairs);
    }
}